// SE_Loss_69423851372618
// MI455X (gfx1250) — compile-verified
//
#include <hip/hip_runtime.h>
#include <hip/hip_bf16.h>

typedef float v2f __attribute__((ext_vector_type(2)));
typedef float v8f __attribute__((ext_vector_type(8)));

#define C_CH    256     // channels
#define CC      16      // channels per staged chunk
#define TP      128     // pixels per block
#define EPS_LN  1e-6f
#define EPS_COS 1e-8f

// ---------------------------------------------------------------------------
// Pass 1: streaming per-pixel stats.
//  - Global -> LDS staging with GLOBAL_LOAD_ASYNC_TO_LDS_B128 (ASYNCcnt,
//    double buffered, 512B per wave-instruction, full-cacheline streams).
//  - Each wave32 owns 16 pixels; five channel reductions (Sa,Sa2,Sb,Sb2,Sab)
//    via V_WMMA_F32_16X16X4_F32 with an all-ones B: D = A(16px x 4ch)*1 + C.
//    A layout per ISA: lanes 0-15 hold K=0,1 (M=lane), lanes 16-31 hold K=2,3.
// ---------------------------------------------------------------------------
__global__ __launch_bounds__(256) void stats_kernel(
    const float* __restrict__ a, const float* __restrict__ b,
    float* __restrict__ score, float* __restrict__ sim,
    int HW, int CHW)
{
    __shared__ float bufA[2][CC * TP];
    __shared__ float bufB[2][CC * TP];

    const int tid  = threadIdx.x;
    const int lane = tid & 31;
    const int wave = tid >> 5;
    const int m    = lane & 15;              // pixel row M within WMMA tile
    const int koff = (lane >> 4) << 1;       // 0 or 2 (K offset per A layout)

    const int  batch = blockIdx.y;
    const int  pix0  = blockIdx.x * TP;
    int valid = HW - pix0; if (valid > TP) valid = TP;   // tail (multiple of 16)
    const bool active = (wave << 4) < valid;
    const long gbase  = (long)batch * CHW + pix0;

    // ---- async staging: thread covers 2 slots of (channel, 4-pixel group) ----
    auto issue = [&](int chunk, int bufsel) {
#pragma unroll
        for (int r = 0; r < 2; ++r) {
            int s  = tid + 256 * r;          // 512 slots = 16 ch x 32 groups
            int ch = s >> 5;
            int pg = (s & 31) << 2;          // pixel group start (x4 floats)
            int pgc = pg; if (pgc > valid - 4) pgc = valid - 4;   // tail clamp
            long ge = gbase + (long)(chunk * CC + ch) * HW + pgc;
            unsigned long long ga = (unsigned long long)(uintptr_t)(a + ge);
            unsigned long long gb = (unsigned long long)(uintptr_t)(b + ge);
            unsigned la = (unsigned)(uintptr_t)&bufA[bufsel][ch * TP + pg];
            unsigned lb = (unsigned)(uintptr_t)&bufB[bufsel][ch * TP + pg];
            asm volatile("global_load_async_to_lds_b128 %0, %1, off"
                         :: "v"(la), "v"(ga) : "memory");
            asm volatile("global_load_async_to_lds_b128 %0, %1, off"
                         :: "v"(lb), "v"(gb) : "memory");
        }
    };

    v8f acc_sa = {}, acc_sa2 = {}, acc_sb = {}, acc_sb2 = {}, acc_sab = {};
    v2f ones; ones.x = 1.0f; ones.y = 1.0f;
    const int widx = (wave << 4) + m;        // pixel within block for this lane

    issue(0, 0);
    const int nch = C_CH / CC;               // 16 chunks
    for (int i = 0; i < nch; ++i) {
        if (i + 1 < nch) {
            issue(i + 1, (i + 1) & 1);
            // current chunk's 4 async ops complete (in-order) when cnt <= 4
            asm volatile("s_wait_asynccnt 0x4" ::: "memory");
        } else {
            asm volatile("s_wait_asynccnt 0x0" ::: "memory");
        }
        __syncthreads();                     // chunk i visible to all waves
        if (active) {
            const float* sA = bufA[i & 1];
            const float* sB = bufB[i & 1];
#pragma unroll
            for (int c = 0; c < CC; c += 4) {
                int i0 = (c + koff) * TP + widx;
                v2f av, bv;
                av.x = sA[i0]; av.y = sA[i0 + TP];
                bv.x = sB[i0]; bv.y = sB[i0 + TP];
                v2f aa = av * av;
                v2f bb = bv * bv;
                v2f ab = av * bv;
                acc_sa  = __builtin_amdgcn_wmma_f32_16x16x4_f32(false, av, false, ones, (short)0, acc_sa,  false, false);
                acc_sa2 = __builtin_amdgcn_wmma_f32_16x16x4_f32(false, aa, false, ones, (short)0, acc_sa2, false, false);
                acc_sb  = __builtin_amdgcn_wmma_f32_16x16x4_f32(false, bv, false, ones, (short)0, acc_sb,  false, false);
                acc_sb2 = __builtin_amdgcn_wmma_f32_16x16x4_f32(false, bb, false, ones, (short)0, acc_sb2, false, false);
                acc_sab = __builtin_amdgcn_wmma_f32_16x16x4_f32(false, ab, false, ones, (short)0, acc_sab, false, false);
            }
        }
        __syncthreads();                     // done reading before overwrite
    }

    // D layout: lanes 0-15 hold rows 0..7 in v[0..7], lanes 16-31 rows 8..15.
    // Columns identical (B was ones) -> lanes 0 and 16 write out.
    if (active && m == 0) {
        const int   rbase = (int)((long)batch * HW) + pix0 + (wave << 4) + ((lane >> 4) << 3);
        const float invC  = 1.0f / (float)C_CH;
        const float invC1 = 1.0f / (float)(C_CH - 1);
#pragma unroll
        for (int r = 0; r < 8; ++r) {
            float sa = acc_sa[r],  sa2 = acc_sa2[r];
            float sb = acc_sb[r],  sb2 = acc_sb2[r];
            float sab = acc_sab[r];
            float csa  = fmaxf(sa2 - sa * sa * invC, 0.0f);
            float csb  = fmaxf(sb2 - sb * sb * invC, 0.0f);
            float cnum = sab - sa * sb * invC;
            // reference semantics: sigma = std(ddof=1) + EPS_LN, clamp norms
            float sda = sqrtf(csa * invC1) + EPS_LN;
            float sdb = sqrtf(csb * invC1) + EPS_LN;
            float na  = sqrtf(csa) / sda;
            float nb  = sqrtf(csb) / sdb;
            float s   = (cnum / (sda * sdb)) / (fmaxf(na, EPS_COS) * fmaxf(nb, EPS_COS));
            score[rbase + r] = sa2;          // raw-bev squared-sum score (>=0)
            sim[rbase + r]   = s;
        }
    }
}

// ---------------------------------------------------------------------------
// Pass 2: per-batch exact k-th-largest threshold via 4-round radix select on
// the fp32 bit pattern (scores >= 0 so uint order == float order).
// ---------------------------------------------------------------------------
__global__ __launch_bounds__(1024) void select_kernel(
    const float* __restrict__ score, unsigned* __restrict__ thr,
    unsigned* __restrict__ need, float* __restrict__ accum,
    int HW, int k)
{
    const int tid = threadIdx.x;
    const int b   = blockIdx.x;
    if (b == 0 && tid == 0) accum[0] = 0.0f;   // reset accumulator for pass 3

    const unsigned* bits = reinterpret_cast<const unsigned*>(score) + (long)b * HW;
    __shared__ unsigned hist[256];
    __shared__ unsigned s_prefix;
    __shared__ unsigned s_remk;
    if (tid == 0) { s_prefix = 0u; s_remk = (unsigned)k; }
    __syncthreads();

    for (int shift = 24; shift >= 0; shift -= 8) {
        for (int i = tid; i < 256; i += blockDim.x) hist[i] = 0u;
        __syncthreads();
        const unsigned prefix = s_prefix;
        const unsigned pmask  = (shift == 24) ? 0u : (0xFFFFFFFFu << (shift + 8));
        for (int i = tid; i < HW; i += blockDim.x) {
            unsigned v = bits[i];
            if ((v & pmask) == prefix)
                atomicAdd(&hist[(v >> shift) & 255u], 1u);
        }
        __syncthreads();
        if (tid == 0) {
            unsigned cum = 0; int sel = 0;
            for (int v = 255; v >= 0; --v) {
                cum += hist[v];
                if (cum >= s_remk) { sel = v; break; }
            }
            s_remk   -= (cum - hist[sel]);     // drop count strictly above sel
            s_prefix  = prefix | ((unsigned)sel << shift);
        }
        __syncthreads();
    }
    if (tid == 0) { thr[b] = s_prefix; need[b] = s_remk; }
}

// ---------------------------------------------------------------------------
// Pass 3: per-batch sum of sim over selected pixels (> threshold, plus the
// required number of ties at the threshold, weighted proportionally).
// ---------------------------------------------------------------------------
__global__ __launch_bounds__(256) void gather_kernel(
    const float* __restrict__ score, const float* __restrict__ sim,
    const unsigned* __restrict__ thr, const unsigned* __restrict__ need,
    float* __restrict__ accum, int HW)
{
    const int tid = threadIdx.x;
    const int b   = blockIdx.x;
    const unsigned T = thr[b];
    const unsigned* bits = reinterpret_cast<const unsigned*>(score) + (long)b * HW;
    const float*    simb = sim + (long)b * HW;

    float sum_gt = 0.0f, sum_eq = 0.0f; unsigned cnt_eq = 0u;
    for (int i = tid; i < HW; i += blockDim.x) {
        unsigned v = bits[i];
        float    s = simb[i];
        if (v > T)       sum_gt += s;
        else if (v == T) { sum_eq += s; cnt_eq++; }
    }

    __shared__ float    r0[256];
    __shared__ float    r1[256];
    __shared__ unsigned r2[256];
    r0[tid] = sum_gt; r1[tid] = sum_eq; r2[tid] = cnt_eq;
    __syncthreads();
    for (int off = 128; off > 0; off >>= 1) {
        if (tid < off) {
            r0[tid] += r0[tid + off];
            r1[tid] += r1[tid + off];
            r2[tid] += r2[tid + off];
        }
        __syncthreads();
    }
    if (tid == 0) {
        float S = r0[0];
        unsigned ce = r2[0];
        if (ce > 0u) S += ((float)need[b] / (float)ce) * r1[0];
        atomicAdd(accum, S);
    }
}

// ---------------------------------------------------------------------------
// Pass 4: scalar finalize (align loss + regularizer). B is tiny (8).
// ---------------------------------------------------------------------------
__global__ void finalize_kernel(const float* __restrict__ accum,
                                const float* __restrict__ dx,
                                const float* __restrict__ dy,
                                const float* __restrict__ dt,
                                float* __restrict__ out, int B, int k)
{
    if (threadIdx.x == 0 && blockIdx.x == 0) {
        float S     = accum[0];
        float align = 1.0f - S / (float)((long)B * (long)k);
        float reg = 0.0f, regt = 0.0f;
        for (int i = 0; i < B; ++i) {
            reg  += dx[i] * dx[i] + dy[i] * dy[i];
            regt += dt[i] * dt[i];
        }
        out[0] = align + 0.1f * (reg / (float)B + regt / (float)B);
    }
}

extern "C" void kernel_launch(void* const* d_in, const int* in_sizes, int n_in,
                              void* d_out, int out_size, void* d_ws, size_t ws_size,
                              hipStream_t stream)
{
    const float* bev = (const float*)d_in[0];
    const float* pri = (const float*)d_in[1];
    const float* dx  = (const float*)d_in[2];
    const float* dy  = (const float*)d_in[3];
    const float* dt  = (const float*)d_in[4];
    float* out = (float*)d_out;

    const int B        = in_sizes[2];          // dx_m has B elements
    const int total    = in_sizes[0];          // B*C*H*W
    const int CHW      = total / B;
    const int HW       = CHW / C_CH;
    const int totalPix = B * HW;
    int k = (int)(0.2 * (double)HW);
    if (k < 1) k = 1;

    // workspace layout
    float*    score = (float*)d_ws;
    float*    sim   = score + totalPix;
    unsigned* thr   = (unsigned*)(sim + totalPix);
    unsigned* need  = thr + B;
    float*    accum = (float*)(need + B);

    dim3 grid1((HW + TP - 1) / TP, B);
    stats_kernel<<<grid1, 256, 0, stream>>>(bev, pri, score, sim, HW, CHW);
    select_kernel<<<B, 1024, 0, stream>>>(score, thr, need, accum, HW, k);
    gather_kernel<<<B, 256, 0, stream>>>(score, sim, thr, need, accum, HW);
    finalize_kernel<<<1, 1, 0, stream>>>(accum, dx, dy, dt, out, B, k);
}